// GCN_56195352100977
// MI455X (gfx1250) — compile-verified
//
#include <hip/hip_runtime.h>

typedef float v2f __attribute__((ext_vector_type(2)));
typedef float v8f __attribute__((ext_vector_type(8)));

#define KC 64
#define LDSS 68   // 64 + 4 pad: row-to-row bank offset of 4 -> conflict-free strided reads

// C[n, NCOLS] = (RELU ? relu(A) : A)[n, K] @ W[NCOLS, K]^T
// One block = 16 rows of A; one wave per 16-column tile of C.
template<int K, int NCOLS, bool RELU_IN>
__global__ __launch_bounds__(32 * (NCOLS / 16))
void gemm_wmma_kernel(const float* __restrict__ A, const float* __restrict__ W,
                      float* __restrict__ C, int n)
{
    constexpr int NT = 32 * (NCOLS / 16);
    __shared__ float sA[16 * LDSS];

    const int rbase = blockIdx.x * 16;
    const int tid   = threadIdx.x;
    const int lane  = tid & 31;
    const int wave  = tid >> 5;
    const int cbase = wave * 16;
    const int mrow  = lane & 15;            // A row / B column held by this lane
    const int khalf = (lane >> 4) << 1;     // 0 (lanes 0-15) or 2 (lanes 16-31)

    v8f acc = {};   // 16x16 f32 accumulator (8 VGPRs)

    for (int kc = 0; kc < K; kc += KC) {
        // Cooperatively stage a 16 x KC chunk of A into LDS (fused ReLU).
        for (int i = tid; i < 16 * (KC / 4); i += NT) {
            const int row = i >> 4;           // 0..15
            const int col = (i & 15) << 2;    // 0..60 step 4
            float4 v = {0.f, 0.f, 0.f, 0.f};
            if (rbase + row < n)
                v = *(const float4*)(A + (size_t)(rbase + row) * K + kc + col);
            if (RELU_IN) {
                v.x = fmaxf(v.x, 0.f); v.y = fmaxf(v.y, 0.f);
                v.z = fmaxf(v.z, 0.f); v.w = fmaxf(v.w, 0.f);
            }
            *(float4*)(&sA[row * LDSS + col]) = v;
        }
        __syncthreads();

        #pragma unroll
        for (int k = 0; k < KC; k += 4) {
            // A fragment 16x4: lane -> row M=lane%16, K = khalf, khalf+1 of this step
            v2f a = *(const v2f*)(&sA[mrow * LDSS + k + khalf]);
            // B fragment 4x16: B[k][nn] = W[cbase+nn][k]; lane holds column N=lane%16
            v2f b = *(const v2f*)(W + (size_t)(cbase + mrow) * K + kc + k + khalf);
            acc = __builtin_amdgcn_wmma_f32_16x16x4_f32(
                /*neg_a=*/false, a, /*neg_b=*/false, b,
                /*c_mod=*/(short)0, acc, /*reuse_a=*/false, /*reuse_b=*/false);
        }
        __syncthreads();
    }

    // Store 16x16 tile: VGPR g holds row M = g + 8*(lane/16), column N = lane%16
    #pragma unroll
    for (int g = 0; g < 8; ++g) {
        const int m = g + ((lane >> 4) << 3);
        if (rbase + m < n)
            C[(size_t)(rbase + m) * NCOLS + cbase + mrow] = acc[g];
    }
}

// out[rows[e]] += vals[e] * h[cols[e]]  -- one wave per edge, F/32 features per lane
template<int F>
__global__ void spmm_scatter_kernel(const float* __restrict__ h,
                                    const int* __restrict__ rows,
                                    const int* __restrict__ cols,
                                    const float* __restrict__ vals,
                                    float* __restrict__ out, int nE)
{
    constexpr int VEC = F / 32;
    const long idx = (long)blockIdx.x * blockDim.x + threadIdx.x;
    const long e   = idx >> 5;
    const int lane = (int)(idx & 31);
    if (e >= (long)nE) return;

    const int   c = cols[e];
    const int   r = rows[e];
    const float v = vals[e];

    const float* hp = h   + (size_t)c * F + lane * VEC;
    float*       op = out + (size_t)r * F + lane * VEC;

    if constexpr (VEC == 4) {
        float4 hv = *(const float4*)hp;
        atomicAdd(op + 0, v * hv.x);
        atomicAdd(op + 1, v * hv.y);
        atomicAdd(op + 2, v * hv.z);
        atomicAdd(op + 3, v * hv.w);
    } else {
        float2 hv = *(const float2*)hp;
        atomicAdd(op + 0, v * hv.x);
        atomicAdd(op + 1, v * hv.y);
    }
}

__global__ void zero_kernel(float4* __restrict__ p, size_t n4)
{
    size_t i      = (size_t)blockIdx.x * blockDim.x + threadIdx.x;
    size_t stride = (size_t)gridDim.x * blockDim.x;
    const float4 z = {0.f, 0.f, 0.f, 0.f};
    for (; i < n4; i += stride) p[i] = z;
}

extern "C" void kernel_launch(void* const* d_in, const int* in_sizes, int n_in,
                              void* d_out, int out_size, void* d_ws, size_t ws_size,
                              hipStream_t stream)
{
    const float* x    = (const float*)d_in[0];
    const int*   rows = (const int*)  d_in[1];
    const int*   cols = (const int*)  d_in[2];
    const float* vals = (const float*)d_in[3];
    const float* W0   = (const float*)d_in[4];   // (128, 512)
    const float* W1   = (const float*)d_in[5];   // (128, 128)
    const float* W2   = (const float*)d_in[6];   // (64, 128)
    float* out = (float*)d_out;

    constexpr int IN = 512, HID = 128, OUT = 64;
    const int n  = in_sizes[0] / IN;   // 100000
    const int nE = in_sizes[1];        // 1600000

    float* hbuf = (float*)d_ws;                  // n * HID floats (also reused for 64-wide h2)
    float* sbuf = hbuf + (size_t)n * HID;        // n * HID floats (SpMM accumulator)

    const int gemmBlocks = (n + 15) / 16;
    const int spmmBlocks = (int)(((long)nE * 32 + 255) / 256);

    // Layer 0: h = x @ W0^T ; s = spmm(h)
    gemm_wmma_kernel<IN, HID, false><<<gemmBlocks, 256, 0, stream>>>(x, W0, hbuf, n);
    zero_kernel<<<2048, 256, 0, stream>>>((float4*)sbuf, (size_t)n * HID / 4);
    spmm_scatter_kernel<HID><<<spmmBlocks, 256, 0, stream>>>(hbuf, rows, cols, vals, sbuf, nE);

    // Layer 1: h = relu(s) @ W1^T ; s = spmm(h)
    gemm_wmma_kernel<HID, HID, true><<<gemmBlocks, 256, 0, stream>>>(sbuf, W1, hbuf, n);
    zero_kernel<<<2048, 256, 0, stream>>>((float4*)sbuf, (size_t)n * HID / 4);
    spmm_scatter_kernel<HID><<<spmmBlocks, 256, 0, stream>>>(hbuf, rows, cols, vals, sbuf, nE);

    // Output layer: h2 = relu(s) @ W2^T ; out = spmm(h2)
    gemm_wmma_kernel<HID, OUT, true><<<gemmBlocks, 128, 0, stream>>>(sbuf, W2, hbuf, n);
    zero_kernel<<<2048, 256, 0, stream>>>((float4*)out, (size_t)n * OUT / 4);
    spmm_scatter_kernel<OUT><<<spmmBlocks, 256, 0, stream>>>(hbuf, rows, cols, vals, out, nE);
}